// DinoGazeSpade_20272245637540
// MI455X (gfx1250) — compile-verified
//
#include <hip/hip_runtime.h>
#include <hip/hip_bf16.h>
#include <math.h>

typedef float v2f __attribute__((ext_vector_type(2)));
typedef float v8f __attribute__((ext_vector_type(8)));

#define NSEG  64
#define CS    384
#define CM    1536
#define HM    128
#define HP    32
#define PIX   1024      // 32*32
#define HI    448
#define BATCH 2
#define KSH   (CS * 9)  // 3456
#define KG    (HM * 9)  // 1152

#define WMMA_F32(A, B, C) \
    __builtin_amdgcn_wmma_f32_16x16x4_f32(false, (A), false, (B), (short)0, (C), false, false)

__device__ __forceinline__ float softplus_f(float x) {
    return fmaxf(x, 0.f) + log1pf(expf(-fabsf(x)));
}

__device__ __forceinline__ int clamp_seg(int s) {
    return s < 0 ? 0 : (s > NSEG - 1 ? NSEG - 1 : s);
}

// ---------------------------------------------------------------------------
// 1) Per-(batch,segment) mean of semantic features over the 32x32 grid.
//    Writes TRANSPOSED layout avgT[b][c][s] -> contiguous B fragments.
// ---------------------------------------------------------------------------
__global__ void seg_stats_kernel(const float* __restrict__ fsem,
                                 const int* __restrict__ segmap,
                                 float* __restrict__ seg_avgT) {
    const int b     = blockIdx.x / 3;
    const int chunk = blockIdx.x % 3;
    const int t     = threadIdx.x;
    __shared__ float s_sum[NSEG * 128];
    __shared__ float s_cnt[NSEG];
    __shared__ int   s_seg[PIX];
    for (int i = t; i < NSEG * 128; i += 256) s_sum[i] = 0.f;
    if (t < NSEG) s_cnt[t] = 0.f;
    for (int p = t; p < PIX; p += 256) {
        int y = p >> 5, x = p & 31;
        s_seg[p] = clamp_seg(segmap[((size_t)b * HI + y * 14) * HI + x * 14]);
    }
    __syncthreads();
    for (int p = t; p < PIX; p += 256) atomicAdd(&s_cnt[s_seg[p]], 1.f);
    const int c  = t & 127;
    const int ph = t >> 7;
    const int cg = chunk * 128 + c;
    for (int it = 0; it < PIX / 2; ++it) {
        int p   = it * 2 + ph;
        float v = fsem[((size_t)(b * CS + cg)) * PIX + p];
        atomicAdd(&s_sum[s_seg[p] * 128 + c], v);
    }
    __syncthreads();
    for (int i = t; i < NSEG * 128; i += 256) {
        int s = i >> 7, cc = i & 127;
        float cnt = s_cnt[s];
        float a   = cnt > 0.f ? s_sum[i] / fmaxf(cnt, 1.f) : 0.f;
        seg_avgT[((size_t)(b * CS + chunk * 128 + cc)) * NSEG + s] = a;
    }
}

// ---------------------------------------------------------------------------
// 2) Per-output-pixel 64-bin segment-weight histogram of the anti-aliased
//    bilinear 448->32 resize (triangle kernel, radius 14, sample_f=14i+6.5).
// ---------------------------------------------------------------------------
__global__ void wmat_kernel(const int* __restrict__ segmap,
                            float* __restrict__ wmat) {
    const int gid = blockIdx.x;
    const int b   = gid >> 10;
    const int p   = gid & 1023;
    const int oy  = p >> 5, ox = p & 31;
    const int t   = threadIdx.x;
    __shared__ float hist[NSEG];
    __shared__ float wy[28], wx[28];
    __shared__ float snorm;
    if (t < NSEG) hist[t] = 0.f;
    if (t < 28) {
        int jy  = 14 * oy - 7 + t;
        float w = 1.f - fabsf(13.5f - (float)t) * (1.f / 14.f);
        wy[t]   = (jy >= 0 && jy < HI) ? w : 0.f;
    }
    if (t >= 32 && t < 60) {
        int tx  = t - 32;
        int jx  = 14 * ox - 7 + tx;
        float w = 1.f - fabsf(13.5f - (float)tx) * (1.f / 14.f);
        wx[tx]  = (jx >= 0 && jx < HI) ? w : 0.f;
    }
    __syncthreads();
    if (t == 0) {
        float sy = 0.f, sx = 0.f;
        for (int i = 0; i < 28; ++i) { sy += wy[i]; sx += wx[i]; }
        snorm = 1.f / (sy * sx);
    }
    __syncthreads();
    for (int tap = t; tap < 28 * 28; tap += 256) {
        int ty = tap / 28, tx = tap - ty * 28;
        float w = wy[ty] * wx[tx];
        if (w > 0.f) {
            int jy = 14 * oy - 7 + ty;
            int jx = 14 * ox - 7 + tx;
            int s  = clamp_seg(segmap[((size_t)b * HI + jy) * HI + jx]);
            atomicAdd(&hist[s], w);
        }
    }
    __syncthreads();
    if (t < NSEG) wmat[(size_t)gid * NSEG + t] = hist[t] * snorm;
}

// ---------------------------------------------------------------------------
// 3) sem_rs[b,c,p] = sum_s W[b,p,s] * avgT[b,c,s] via f32 WMMA.
//    One wave -> 16x32 tile (two N-tiles sharing A).
// ---------------------------------------------------------------------------
__global__ void semrs_wmma_kernel(const float* __restrict__ wmat,
                                  const float* __restrict__ seg_avgT,
                                  float* __restrict__ semrs) {
    const int wave = (blockIdx.x * blockDim.x + threadIdx.x) >> 5;
    const int lane = threadIdx.x & 31;
    const int NP = CS / 32, PT = PIX / 16;   // 12 n-pairs
    const int np = wave % NP;
    const int pt = (wave / NP) % PT;
    const int b  = wave / (NP * PT);
    const int col   = lane & 15;
    const int khalf = (lane >> 4) << 1;
    const int m  = pt * 16 + col;
    const int n0 = np * 32 + col;
    const int n1 = n0 + 16;
    const float* Arow = wmat + ((size_t)(b * PIX + m)) * NSEG + khalf;
    const float* B0   = seg_avgT + ((size_t)(b * CS + n0)) * NSEG + khalf;
    const float* B1   = seg_avgT + ((size_t)(b * CS + n1)) * NSEG + khalf;
    v8f acc0 = {}, acc1 = {};
    for (int k0 = 0; k0 < NSEG; k0 += 4) {
        v2f av, b0v, b1v;
        av.x  = Arow[k0];  av.y  = Arow[k0 + 1];
        b0v.x = B0[k0];    b0v.y = B0[k0 + 1];
        b1v.x = B1[k0];    b1v.y = B1[k0 + 1];
        acc0 = WMMA_F32(av, b0v, acc0);
        acc1 = WMMA_F32(av, b1v, acc1);
    }
    const int hi = lane >> 4;
#pragma unroll
    for (int v = 0; v < 8; ++v) {
        int row = pt * 16 + v + hi * 8;   // D: VGPR v -> M = v + 8*hi
        semrs[((size_t)(b * CS + n0)) * PIX + row] = acc0[v];
        semrs[((size_t)(b * CS + n1)) * PIX + row] = acc1[v];
    }
}

// ---------------------------------------------------------------------------
// Weight permutation: w[co][ci][3][3] -> wp[co][tap][ci] (contiguous B pairs).
// ---------------------------------------------------------------------------
__global__ void wperm_kernel(const float* __restrict__ w,
                             float* __restrict__ wp,
                             int total, int Cin) {
    int idx = blockIdx.x * 256 + threadIdx.x;   // over Cout*Cin*9
    if (idx >= total) return;
    int tap = idx % 9;
    int ci  = (idx / 9) % Cin;
    int co  = idx / (9 * Cin);
    wp[((size_t)co * 9 + tap) * Cin + ci] = w[idx];
}

// ---------------------------------------------------------------------------
// 4) Shared SPADE conv: h = relu(conv3x3(sem_rs, w)+b). Cin=384, Cout=128.
//    Tap-major implicit GEMM (k' = tap*Cin + ci); y is wave-uniform so the ky
//    check is a uniform branch; kx edge handled by a hoisted multiply mask.
//    FOUR N-tiles per wave share each A fragment: 2 A + 4 b64 per 4 WMMAs.
// ---------------------------------------------------------------------------
__global__ void conv_sh_wmma_kernel(const float* __restrict__ semrs,
                                    const float* __restrict__ wp,   // [Cout][9][CS]
                                    const float* __restrict__ bias,
                                    float* __restrict__ hout) {
    const int wave = (blockIdx.x * blockDim.x + threadIdx.x) >> 5;
    const int lane = threadIdx.x & 31;
    const int NQ = HM / 64, PT = PIX / 16;    // 2 n-quads
    const int nq = wave % NQ;
    const int pt = (wave / NQ) % PT;
    const int b  = wave / (NQ * PT);
    const int col   = lane & 15;
    const int khalf = (lane >> 4) << 1;
    const int m = pt * 16 + col;
    const int y = m >> 5, x = m & 31;         // y uniform across the wave
    const int nb = nq * 64 + col;
    const float* in = semrs + (size_t)b * CS * PIX + (size_t)khalf * PIX;
    const float* w0 = wp + (size_t)(nb)      * KSH + khalf;
    const float* w1 = wp + (size_t)(nb + 16) * KSH + khalf;
    const float* w2 = wp + (size_t)(nb + 32) * KSH + khalf;
    const float* w3 = wp + (size_t)(nb + 48) * KSH + khalf;
    v8f a0 = {}, a1 = {}, a2 = {}, a3 = {};
    for (int ky = 0; ky < 3; ++ky) {
        int iy = y + ky - 1;
        if (iy < 0 || iy >= HP) continue;     // uniform branch
        for (int kx = 0; kx < 3; ++kx) {
            int ix = x + kx - 1;
            float xm = (ix >= 0 && ix < HP) ? 1.f : 0.f;
            int ixc = ix < 0 ? 0 : (ix > HP - 1 ? HP - 1 : ix);
            const float* pin = in + (size_t)iy * HP + ixc;
            const int tof = (ky * 3 + kx) * CS;
            for (int ci = 0; ci < CS; ci += 4) {
                v2f av, b0v, b1v, b2v, b3v;
                av.x  = pin[(size_t)ci * PIX] * xm;
                av.y  = pin[(size_t)(ci + 1) * PIX] * xm;
                b0v.x = w0[tof + ci];  b0v.y = w0[tof + ci + 1];
                b1v.x = w1[tof + ci];  b1v.y = w1[tof + ci + 1];
                b2v.x = w2[tof + ci];  b2v.y = w2[tof + ci + 1];
                b3v.x = w3[tof + ci];  b3v.y = w3[tof + ci + 1];
                a0 = WMMA_F32(av, b0v, a0);
                a1 = WMMA_F32(av, b1v, a1);
                a2 = WMMA_F32(av, b2v, a2);
                a3 = WMMA_F32(av, b3v, a3);
            }
        }
    }
    const int hi = lane >> 4;
    const float bn0 = bias[nb], bn1 = bias[nb + 16];
    const float bn2 = bias[nb + 32], bn3 = bias[nb + 48];
#pragma unroll
    for (int v = 0; v < 8; ++v) {
        int row = pt * 16 + v + hi * 8;
        hout[((size_t)(b * HM + nb))      * PIX + row] = fmaxf(a0[v] + bn0, 0.f);
        hout[((size_t)(b * HM + nb + 16)) * PIX + row] = fmaxf(a1[v] + bn1, 0.f);
        hout[((size_t)(b * HM + nb + 32)) * PIX + row] = fmaxf(a2[v] + bn2, 0.f);
        hout[((size_t)(b * HM + nb + 48)) * PIX + row] = fmaxf(a3[v] + bn3, 0.f);
    }
}

// ---------------------------------------------------------------------------
// 5) Fused gamma/beta convs (Cin=128, tap-major) + SPADE apply:
//    y = (x-mu)*rstd*(1+gamma)+beta.
//    TWO N-tiles x {gamma,beta} = 4 accumulators share each A fragment.
// ---------------------------------------------------------------------------
__global__ void spade_wmma_kernel(const float* __restrict__ hsh,
                                  const float* __restrict__ wpg,   // [N][9][HM]
                                  const float* __restrict__ bg,
                                  const float* __restrict__ wpbe,  // [N][9][HM]
                                  const float* __restrict__ bbe,
                                  const float* __restrict__ xsrc,
                                  const float* __restrict__ mr,    // [B][2]
                                  float* __restrict__ yout,
                                  int Nch) {
    const int wave = (blockIdx.x * blockDim.x + threadIdx.x) >> 5;
    const int lane = threadIdx.x & 31;
    const int NPAIR = (Nch + 31) >> 5, PT = PIX / 16;
    const int np = wave % NPAIR;
    const int pt = (wave / NPAIR) % PT;
    const int b  = wave / (NPAIR * PT);
    const int col   = lane & 15;
    const int khalf = (lane >> 4) << 1;
    const int m = pt * 16 + col;
    const int y = m >> 5, x = m & 31;         // y uniform across the wave
    const int n0 = np * 32 + col;
    const int n1 = n0 + 16;
    const bool v0 = (n0 < Nch), v1 = (n1 < Nch);
    const int nc0 = v0 ? n0 : 0, nc1 = v1 ? n1 : 0;
    const float* in  = hsh + (size_t)b * HM * PIX + (size_t)khalf * PIX;
    const float* wg0 = wpg  + (size_t)nc0 * KG + khalf;
    const float* wg1 = wpg  + (size_t)nc1 * KG + khalf;
    const float* wb0 = wpbe + (size_t)nc0 * KG + khalf;
    const float* wb1 = wpbe + (size_t)nc1 * KG + khalf;
    v8f ag0 = {}, ab0 = {}, ag1 = {}, ab1 = {};
    for (int ky = 0; ky < 3; ++ky) {
        int iy = y + ky - 1;
        if (iy < 0 || iy >= HP) continue;     // uniform branch
        for (int kx = 0; kx < 3; ++kx) {
            int ix = x + kx - 1;
            float xm = (ix >= 0 && ix < HP) ? 1.f : 0.f;
            int ixc = ix < 0 ? 0 : (ix > HP - 1 ? HP - 1 : ix);
            const float* pin = in + (size_t)iy * HP + ixc;
            const int tof = (ky * 3 + kx) * HM;
            for (int ci = 0; ci < HM; ci += 4) {
                v2f av, g0v, g1v, e0v, e1v;
                av.x = pin[(size_t)ci * PIX] * xm;
                av.y = pin[(size_t)(ci + 1) * PIX] * xm;
                g0v.x = wg0[tof + ci];  g0v.y = wg0[tof + ci + 1];
                e0v.x = wb0[tof + ci];  e0v.y = wb0[tof + ci + 1];
                g1v.x = wg1[tof + ci];  g1v.y = wg1[tof + ci + 1];
                e1v.x = wb1[tof + ci];  e1v.y = wb1[tof + ci + 1];
                ag0 = WMMA_F32(av, g0v, ag0);
                ab0 = WMMA_F32(av, e0v, ab0);
                ag1 = WMMA_F32(av, g1v, ag1);
                ab1 = WMMA_F32(av, e1v, ab1);
            }
        }
    }
    const int hi     = lane >> 4;
    const float mu   = mr[b * 2 + 0];
    const float rstd = mr[b * 2 + 1];
    if (v0) {
        const float gb = bg[n0], eb = bbe[n0];
        const float* xb = xsrc + ((size_t)(b * Nch + n0)) * PIX;
        float*       yo = yout + ((size_t)(b * Nch + n0)) * PIX;
#pragma unroll
        for (int v = 0; v < 8; ++v) {
            int row  = pt * 16 + v + hi * 8;
            float xn = (xb[row] - mu) * rstd;
            yo[row]  = xn * (1.f + ag0[v] + gb) + (ab0[v] + eb);
        }
    }
    if (v1) {
        const float gb = bg[n1], eb = bbe[n1];
        const float* xb = xsrc + ((size_t)(b * Nch + n1)) * PIX;
        float*       yo = yout + ((size_t)(b * Nch + n1)) * PIX;
#pragma unroll
        for (int v = 0; v < 8; ++v) {
            int row  = pt * 16 + v + hi * 8;
            float xn = (xb[row] - mu) * rstd;
            yo[row]  = xn * (1.f + ag1[v] + gb) + (ab1[v] + eb);
        }
    }
}

// ---------------------------------------------------------------------------
// 6) Per-batch LayerNorm statistics -> (mu, rstd).
// ---------------------------------------------------------------------------
__global__ void ln_stats_kernel(const float* __restrict__ xsrc, int nPerB,
                                float* __restrict__ mr) {
    const int b = blockIdx.x;
    const int t = threadIdx.x;   // 256
    const float* x = xsrc + (size_t)b * nPerB;
    double s = 0.0, s2 = 0.0;
    for (int i = t; i < nPerB; i += 256) {
        double v = (double)x[i];
        s += v; s2 += v * v;
    }
    __shared__ double sh[256], sh2[256];
    sh[t] = s; sh2[t] = s2;
    __syncthreads();
    for (int o = 128; o > 0; o >>= 1) {
        if (t < o) { sh[t] += sh[t + o]; sh2[t] += sh2[t + o]; }
        __syncthreads();
    }
    if (t == 0) {
        double mu  = sh[0] / nPerB;
        double var = sh2[0] / nPerB - mu * mu;
        mr[b * 2 + 0] = (float)mu;
        mr[b * 2 + 1] = (float)(1.0 / sqrt(var + 1e-12));
    }
}

// ---------------------------------------------------------------------------
// 7) 1x1 conv + softplus.  One thread per (b, co, pixel); coalesced in p.
// ---------------------------------------------------------------------------
__global__ void conv1x1_softplus_kernel(const float* __restrict__ yin,
                                        const float* __restrict__ w,
                                        const float* __restrict__ bias,
                                        float* __restrict__ out,
                                        int Cin, int Cout) {
    int idx = blockIdx.x * blockDim.x + threadIdx.x;
    if (idx >= BATCH * Cout * PIX) return;
    int p = idx & 1023;
    int j = (idx >> 10) % Cout;
    int b = idx / (Cout * PIX);
    const float* yb = yin + (size_t)b * Cin * PIX + p;
    const float* wj = w + (size_t)j * Cin;
    float s = bias[j];
    for (int c = 0; c < Cin; ++c)
        s += yb[(size_t)c * PIX] * wj[c];
    out[idx] = softplus_f(s);
}

// ---------------------------------------------------------------------------
extern "C" void kernel_launch(void* const* d_in, const int* in_sizes, int n_in,
                              void* d_out, int out_size, void* d_ws, size_t ws_size,
                              hipStream_t stream) {
    (void)in_sizes; (void)n_in; (void)out_size; (void)ws_size;
    const float* x_main = (const float*)d_in[0];
    const float* f_sem  = (const float*)d_in[1];
    const float* w_sh0 = (const float*)d_in[2];
    const float* b_sh0 = (const float*)d_in[3];
    const float* w_g0  = (const float*)d_in[4];
    const float* b_g0  = (const float*)d_in[5];
    const float* w_be0 = (const float*)d_in[6];
    const float* b_be0 = (const float*)d_in[7];
    const float* w_sh1 = (const float*)d_in[8];
    const float* b_sh1 = (const float*)d_in[9];
    const float* w_g1  = (const float*)d_in[10];
    const float* b_g1  = (const float*)d_in[11];
    const float* w_be1 = (const float*)d_in[12];
    const float* b_be1 = (const float*)d_in[13];
    const float* w_sh2 = (const float*)d_in[14];
    const float* b_sh2 = (const float*)d_in[15];
    const float* w_g2  = (const float*)d_in[16];
    const float* b_g2  = (const float*)d_in[17];
    const float* w_be2 = (const float*)d_in[18];
    const float* b_be2 = (const float*)d_in[19];
    const float* w_c0  = (const float*)d_in[20];
    const float* bias0 = (const float*)d_in[21];
    const float* w_c1  = (const float*)d_in[22];
    const float* bias1 = (const float*)d_in[23];
    const float* w_c2  = (const float*)d_in[24];
    const float* bias2 = (const float*)d_in[25];
    const int*   segmap = (const int*)d_in[26];

    // Workspace layout (~34 MB of floats)
    float* ws       = (float*)d_ws;
    float* seg_avgT = ws;                                    // 2*384*64
    float* wmat     = seg_avgT + BATCH * CS * NSEG;          // 2*1024*64
    float* semrs    = wmat + BATCH * PIX * NSEG;             // 2*384*1024
    float* hsh      = semrs + (size_t)BATCH * CS * PIX;      // 2*128*1024 (x3 reuse)
    float* ybig     = hsh + BATCH * HM * PIX;                // 2*1536*1024
    float* h8       = ybig + (size_t)BATCH * CM * PIX;       // 2*8*1024
    float* y8       = h8 + BATCH * 8 * PIX;
    float* h16      = y8 + BATCH * 8 * PIX;                  // 2*16*1024
    float* y16      = h16 + BATCH * 16 * PIX;
    float* mr       = y16 + BATCH * 16 * PIX;                // 12
    float* wp_sh    = mr + 16;                               // 128*3456 (x3 reuse)
    float* wp_g     = wp_sh + (size_t)HM * KSH;              // 1536*1152 (x3 reuse)
    float* wp_be    = wp_g + (size_t)CM * KG;                // 1536*1152 (x3 reuse)

    // Stage A: segment averages + resize weight field + sem_rs
    seg_stats_kernel<<<BATCH * 3, 256, 0, stream>>>(f_sem, segmap, seg_avgT);
    wmat_kernel<<<BATCH * PIX, 256, 0, stream>>>(segmap, wmat);
    {
        int waves = BATCH * (PIX / 16) * (CS / 32);          // 1536
        semrs_wmma_kernel<<<waves / 4, 128, 0, stream>>>(wmat, seg_avgT, semrs);
    }

    const int shWaves = BATCH * (PIX / 16) * (HM / 64);      // 256
    const int shTot   = HM * KSH;

    // ---- SPADE block 0 (N = 1536) ----
    ln_stats_kernel<<<BATCH, 256, 0, stream>>>(x_main, CM * PIX, mr);
    wperm_kernel<<<(shTot + 255) / 256, 256, 0, stream>>>(w_sh0, wp_sh, shTot, CS);
    conv_sh_wmma_kernel<<<shWaves / 4, 128, 0, stream>>>(semrs, wp_sh, b_sh0, hsh);
    {
        int gTot = CM * KG;
        wperm_kernel<<<(gTot + 255) / 256, 256, 0, stream>>>(w_g0, wp_g, gTot, HM);
        wperm_kernel<<<(gTot + 255) / 256, 256, 0, stream>>>(w_be0, wp_be, gTot, HM);
        int waves = BATCH * (PIX / 16) * (CM / 32);          // 6144
        spade_wmma_kernel<<<waves / 4, 128, 0, stream>>>(hsh, wp_g, b_g0, wp_be, b_be0,
                                                         x_main, mr, ybig, CM);
    }
    {
        int n1 = BATCH * 8 * PIX;
        conv1x1_softplus_kernel<<<(n1 + 255) / 256, 256, 0, stream>>>(
            ybig, w_c0, bias0, h8, CM, 8);
    }

    // ---- SPADE block 1 (N = 8) ----
    ln_stats_kernel<<<BATCH, 256, 0, stream>>>(h8, 8 * PIX, mr + 4);
    wperm_kernel<<<(shTot + 255) / 256, 256, 0, stream>>>(w_sh1, wp_sh, shTot, CS);
    conv_sh_wmma_kernel<<<shWaves / 4, 128, 0, stream>>>(semrs, wp_sh, b_sh1, hsh);
    {
        int gTot = 8 * KG;
        wperm_kernel<<<(gTot + 255) / 256, 256, 0, stream>>>(w_g1, wp_g, gTot, HM);
        wperm_kernel<<<(gTot + 255) / 256, 256, 0, stream>>>(w_be1, wp_be, gTot, HM);
        int waves = BATCH * (PIX / 16) * 1;                  // 128 (NPAIR=1)
        spade_wmma_kernel<<<waves / 4, 128, 0, stream>>>(hsh, wp_g, b_g1, wp_be, b_be1,
                                                         h8, mr + 4, y8, 8);
    }
    {
        int n1 = BATCH * 16 * PIX;
        conv1x1_softplus_kernel<<<(n1 + 255) / 256, 256, 0, stream>>>(
            y8, w_c1, bias1, h16, 8, 16);
    }

    // ---- SPADE block 2 (N = 16) ----
    ln_stats_kernel<<<BATCH, 256, 0, stream>>>(h16, 16 * PIX, mr + 8);
    wperm_kernel<<<(shTot + 255) / 256, 256, 0, stream>>>(w_sh2, wp_sh, shTot, CS);
    conv_sh_wmma_kernel<<<shWaves / 4, 128, 0, stream>>>(semrs, wp_sh, b_sh2, hsh);
    {
        int gTot = 16 * KG;
        wperm_kernel<<<(gTot + 255) / 256, 256, 0, stream>>>(w_g2, wp_g, gTot, HM);
        wperm_kernel<<<(gTot + 255) / 256, 256, 0, stream>>>(w_be2, wp_be, gTot, HM);
        int waves = BATCH * (PIX / 16) * 1;                  // 128 (NPAIR=1)
        spade_wmma_kernel<<<waves / 4, 128, 0, stream>>>(hsh, wp_g, b_g2, wp_be, b_be2,
                                                         h16, mr + 8, y16, 16);
    }
    {
        int n1 = BATCH * 1 * PIX;
        conv1x1_softplus_kernel<<<(n1 + 255) / 256, 256, 0, stream>>>(
            y16, w_c2, bias2, (float*)d_out, 16, 1);
    }
}